// CorrBlock2_14199161880886
// MI455X (gfx1250) — compile-verified
//
#include <hip/hip_runtime.h>
#include <hip/hip_bf16.h>

typedef __attribute__((ext_vector_type(2))) float v2f;
typedef __attribute__((ext_vector_type(8))) float v8f;

// Problem constants from the reference trace:
// b=2, n_p=16384, K=128, ids=3 (adaptive path since 2<=3<10), dynamic_k=18
#define B_    2
#define NP_   16384
#define KN_   128
#define DK_   18
#define C1_   64
#define CIN1_ 27
#define CINP_ 28           // K padded to multiple of 4 for 16x16x4 WMMA
#define WPB_  8            // waves (=points) per block in the KNN kernel
#define EPS_  1e-8f
#define GEPS_ 1e-5f
#define FINF_ 3.0e38f

// ---- workspace layout (float offsets) -------------------------------------
#define OFF_ATOM  0                                  // r1max[2], r2max[2] as uint bit-max
#define OFF_XSTAT 64                                 // mean[16], rstd[16]
#define OFF_YSTAT 96                                 // mean[16], rstd[16]
#define OFF_YPART 128                                // B*(NP/WPB)*16 deterministic partials
#define OFF_FEAT  (OFF_YPART + B_*(NP_/WPB_)*16)     // B*28*NP feature matrix (ch-major)
#define OFF_KNN   (OFF_FEAT + B_*CINP_*NP_)          // B*NP*DK*3 (corr,dx,dy)
#define OFF_X1    (OFF_KNN + B_*NP_*DK_*3)           // B*64*NP conv1 output
#define OFF_YMAX  (OFF_X1 + B_*C1_*NP_)              // B*64*NP maxpooled y

// ---- K1: per-batch max of |f1| and |r2_perp| (order-independent bit max) --
__global__ void k1_rmax(const float* __restrict__ adf, float* __restrict__ atom) {
  int b = blockIdx.y;
  int p = blockIdx.x * blockDim.x + threadIdx.x;
  const float* f1p = adf + (size_t)((2 * B_ + b) * NP_ + p) * 2;
  const float* f2p = adf + (size_t)((1 * B_ + b) * NP_ + p) * 2;
  float f1x = f1p[0], f1y = f1p[1], f2x = f2p[0], f2y = f2p[1];
  float r1n = sqrtf(f1x * f1x + f1y * f1y);
  float f2n = sqrtf(f2x * f2x + f2y * f2y);
  float cos1 = (f1x * f2x + f1y * f2y) / (fmaxf(r1n, EPS_) * fmaxf(f2n, EPS_));
  float t = cos1 * f2n / r1n;
  float r2x = f2x - t * f1x, r2y = f2y - t * f1y;
  float r2n = sqrtf(r2x * r2x + r2y * r2y);
  __shared__ float s1[256], s2[256];
  int tid = threadIdx.x;
  s1[tid] = r1n; s2[tid] = r2n;
  __syncthreads();
  for (int off = 128; off >= 1; off >>= 1) {
    if (tid < off) {
      s1[tid] = fmaxf(s1[tid], s1[tid + off]);
      s2[tid] = fmaxf(s2[tid], s2[tid + off]);
    }
    __syncthreads();
  }
  if (tid == 0) {
    unsigned* u = (unsigned*)atom;       // norms >= 0 => uint order == float order
    atomicMax(&u[b],     __float_as_uint(s1[0]));
    atomicMax(&u[2 + b], __float_as_uint(s2[0]));
  }
}

// ---- K2a: per-point 3-level 3x3 binned correlation features ---------------
__global__ void k2a_feat(const float* __restrict__ coords, const float* __restrict__ adf,
                         const float* __restrict__ corr, const float* __restrict__ xy2,
                         const int* __restrict__ scale_p, const float* __restrict__ atom,
                         float* __restrict__ feat) {
  int b = blockIdx.y;
  int p = blockIdx.x * blockDim.x + threadIdx.x;
  size_t pbase = (size_t)(b * NP_ + p);
  float cx = coords[pbase * 2 + 0], cy = coords[pbase * 2 + 1];
  const float* f1p = adf + (size_t)((2 * B_ + b) * NP_ + p) * 2;
  const float* f2p = adf + (size_t)((1 * B_ + b) * NP_ + p) * 2;
  float f1x = f1p[0], f1y = f1p[1], f2x = f2p[0], f2y = f2p[1];
  float r1n = sqrtf(f1x * f1x + f1y * f1y);
  float f2n = sqrtf(f2x * f2x + f2y * f2y);
  float cos1 = (f1x * f2x + f1y * f2y) / (fmaxf(r1n, EPS_) * fmaxf(f2n, EPS_));
  float t = cos1 * f2n / r1n;
  float r2x = f2x - t * f1x, r2y = f2y - t * f1y;
  float r2n = sqrtf(r2x * r2x + r2y * r2y);
  const unsigned* u = (const unsigned*)atom;
  float r1max = __uint_as_float(u[b]);
  float r2max = __uint_as_float(u[2 + b]);
  float u1x = f1x / r1n, u1y = f1y / r1n;
  float u2x = r2x / r2n, u2y = r2y / r2n;
  float scl = (float)scale_p[0];

  float ax[3], ay[3], ai[3], bx[3], by[3], bi[3];
#pragma unroll
  for (int i = 0; i < 3; ++i) {
    float r = scl * (float)(1 << i);
    float fac1 = (0.125f * r1n / r1max + 0.125f * cos1 + 1.0f) * r;
    float fac2 = (0.125f * r2n / r2max + 0.9f) * r;
    ax[i] = u1x * fac1; ay[i] = u1y * fac1;
    bx[i] = u2x * fac2; by[i] = u2y * fac2;
    ai[i] = 1.0f / sqrtf(ax[i] * ax[i] + ay[i] * ay[i]);
    bi[i] = 1.0f / sqrtf(bx[i] * bx[i] + by[i] * by[i]);
  }
  float sums[27], cnts[27];
#pragma unroll
  for (int i = 0; i < 27; ++i) { sums[i] = 0.f; cnts[i] = 0.f; }

  const float* xyp = xy2 + pbase * KN_ * 2;
  const float* crp = corr + pbase * KN_;
  for (int k = 0; k < KN_; ++k) {
    float dx = xyp[k * 2 + 0] - cx, dy = xyp[k * 2 + 1] - cy;
    float ck = crp[k];
#pragma unroll
    for (int i = 0; i < 3; ++i) {
      float d0 = rintf((dx * ax[i] + dy * ay[i]) * ai[i]);   // RTE == jnp.round
      float d1 = rintf((dx * bx[i] + dy * by[i]) * bi[i]);
#pragma unroll
      for (int bb = 0; bb < 9; ++bb) {   // static-index accumulate (no scratch)
        bool hit = (d0 == (float)(bb / 3 - 1)) && (d1 == (float)(bb % 3 - 1));
        sums[i * 9 + bb] += hit ? ck : 0.f;
        cnts[i * 9 + bb] += hit ? 1.f : 0.f;
      }
    }
  }
#pragma unroll
  for (int c = 0; c < 27; ++c) {
    float cnt = fminf(fmaxf(cnts[c], 1.0f), (float)NP_);
    feat[((size_t)(b * CINP_ + c)) * NP_ + p] = sums[c] / cnt;
  }
  feat[((size_t)(b * CINP_ + 27)) * NP_ + p] = 0.f;  // zero pad row -> K=28
}

// ---- K2b: wave-per-point KNN top-18 + deterministic y GroupNorm partials --
__global__ void k2b_knn(const float* __restrict__ coords, const float* __restrict__ corr,
                        const float* __restrict__ xy2, const float* __restrict__ wk,
                        const float* __restrict__ bk,
                        float* __restrict__ knn, float* __restrict__ ypart) {
  __shared__ float sred[WPB_][16];
  int lane = threadIdx.x;
  int w = threadIdx.y;
  int b = blockIdx.y;
  int p = blockIdx.x * WPB_ + w;
  size_t pbase = (size_t)(b * NP_ + p);
  float cx = coords[pbase * 2 + 0], cy = coords[pbase * 2 + 1];
  float d[4], cr[4], dxv[4], dyv[4];
#pragma unroll
  for (int s = 0; s < 4; ++s) {
    int k = lane + s * 32;
    size_t kb = pbase * KN_ + k;
    float xx = xy2[kb * 2 + 0] - cx;
    float yy = xy2[kb * 2 + 1] - cy;
    cr[s] = corr[kb]; dxv[s] = xx; dyv[s] = yy;
    d[s] = xx * xx + yy * yy;
  }
  int cA = lane, cB = lane + 32;   // each lane owns 2 of the 64 output channels
  float wA0 = wk[cA * 3 + 0], wA1 = wk[cA * 3 + 1], wA2 = wk[cA * 3 + 2], bA = bk[cA];
  float wB0 = wk[cB * 3 + 0], wB1 = wk[cB * 3 + 1], wB2 = wk[cB * 3 + 2], bB = bk[cB];
  float sA = 0.f, qA = 0.f, sB = 0.f, qB = 0.f;
  float* kout = knn + pbase * (DK_ * 3);

  for (int j = 0; j < DK_; ++j) {
    float ld = d[0]; int lk = lane;
#pragma unroll
    for (int s = 1; s < 4; ++s) {   // strict < keeps smallest k on ties
      int k = lane + s * 32;
      if (d[s] < ld) { ld = d[s]; lk = k; }
    }
#pragma unroll
    for (int off = 16; off >= 1; off >>= 1) {   // butterfly -> all lanes get argmin
      float od = __shfl_xor(ld, off, 32);
      int   ok = __shfl_xor(lk, off, 32);
      if (od < ld || (od == ld && ok < lk)) { ld = od; lk = ok; }  // jax tie-break
    }
    int slot = lk >> 5, owner = lk & 31;        // uniform across the wave
    float cc = (slot == 0) ? cr[0]  : (slot == 1) ? cr[1]  : (slot == 2) ? cr[2]  : cr[3];
    float xx = (slot == 0) ? dxv[0] : (slot == 1) ? dxv[1] : (slot == 2) ? dxv[2] : dxv[3];
    float yy = (slot == 0) ? dyv[0] : (slot == 1) ? dyv[1] : (slot == 2) ? dyv[2] : dyv[3];
    float wc = __shfl(cc, owner, 32);
    float wx = __shfl(xx, owner, 32);
    float wy = __shfl(yy, owner, 32);
    bool mine = (lane == owner);
    d[0] = (mine && slot == 0) ? FINF_ : d[0];
    d[1] = (mine && slot == 1) ? FINF_ : d[1];
    d[2] = (mine && slot == 2) ? FINF_ : d[2];
    d[3] = (mine && slot == 3) ? FINF_ : d[3];
    if (lane == 0) { kout[j * 3 + 0] = wc; kout[j * 3 + 1] = wx; kout[j * 3 + 2] = wy; }
    float yAv = wA0 * wc + wA1 * wx + wA2 * wy + bA;   // conv_k for this lane's channels
    float yBv = wB0 * wc + wB1 * wx + wB2 * wy + bB;
    sA += yAv; qA += yAv * yAv; sB += yBv; qB += yBv * yBv;
  }
  // reduce within 8-lane channel groups (same GN group)
#pragma unroll
  for (int off = 4; off >= 1; off >>= 1) {
    sA += __shfl_xor(sA, off, 32); qA += __shfl_xor(qA, off, 32);
    sB += __shfl_xor(sB, off, 32); qB += __shfl_xor(qB, off, 32);
  }
  if ((lane & 7) == 0) {
    int g = lane >> 3;                       // groups 0..3 (chan<32), 4..7 (chan>=32)
    sred[w][g]      = sA; sred[w][8 + g]  = qA;
    sred[w][4 + g]  = sB; sred[w][12 + g] = qB;
  }
  __syncthreads();
  if (w == 0 && lane < 16) {                 // fixed-order -> deterministic
    float acc = 0.f;
#pragma unroll
    for (int ww = 0; ww < WPB_; ++ww) acc += sred[ww][lane];
    ypart[((size_t)(b * (NP_ / WPB_)) + blockIdx.x) * 16 + lane] = acc;
  }
}

// ---- K6: finalize y GroupNorm stats (fixed-order serial sum) --------------
__global__ void k6_ystats(const float* __restrict__ ypart, float* __restrict__ ystat) {
  int t = threadIdx.x;
  if (t >= 16) return;
  int b = t >> 3, g = t & 7;
  const int NB = NP_ / WPB_;
  float s = 0.f, q = 0.f;
  for (int blk = 0; blk < NB; ++blk) {
    const float* pp = ypart + ((size_t)(b * NB) + blk) * 16;
    s += pp[g]; q += pp[8 + g];
  }
  float cnt = 8.0f * NP_ * DK_;
  float mean = s / cnt;
  float var = q / cnt - mean * mean;
  ystat[t] = mean;
  ystat[16 + t] = rsqrtf(var + GEPS_);
}

// ---- f32 WMMA helper (V_WMMA_F32_16X16X4_F32) -----------------------------
__device__ __forceinline__ v8f wmma4(v2f a, v2f bm, v8f c) {
  return __builtin_amdgcn_wmma_f32_16x16x4_f32(false, a, false, bm, (short)0, c,
                                               false, false);
}

// ---- K3: X1 = W1[64x27] * F[28xNP] + b1 (WMMA) ----------------------------
__global__ void k3_gemm1(const float* __restrict__ w1, const float* __restrict__ b1,
                         const float* __restrict__ feat, float* __restrict__ x1) {
  int lane = threadIdx.x & 31;
  int mt = threadIdx.x >> 5;                 // 4 waves -> 4 M-tiles of 16 channels
  int b = blockIdx.y;
  int col = blockIdx.x * 16 + (lane & 15);
  int half = lane >> 4;
  int m = mt * 16 + (lane & 15);
  v8f acc = {};
#pragma unroll
  for (int kk = 0; kk < 7; ++kk) {           // K = 28
    int k0 = kk * 4 + 2 * half;
    v2f a, bm;
    a.x = (k0     < CIN1_) ? w1[m * CIN1_ + k0]     : 0.f;
    a.y = (k0 + 1 < CIN1_) ? w1[m * CIN1_ + k0 + 1] : 0.f;
    const float* fb = feat + ((size_t)(b * CINP_ + k0)) * NP_ + col;
    bm.x = fb[0]; bm.y = fb[NP_];
    acc = wmma4(a, bm, acc);
  }
#pragma unroll
  for (int v = 0; v < 8; ++v) {
    int mm = mt * 16 + v + 8 * half;
    x1[((size_t)(b * C1_ + mm)) * NP_ + col] = acc[v] + b1[mm];
  }
}

// ---- K4: x GroupNorm stats (one block per (b,g), fixed tree) --------------
__global__ void k4_xstats(const float* __restrict__ x1, float* __restrict__ xstat) {
  int bg = blockIdx.x;
  int b = bg >> 3, g = bg & 7;
  const float* base = x1 + ((size_t)(b * C1_ + g * 8)) * NP_;
  float s = 0.f, q = 0.f;
  for (int i = threadIdx.x; i < 8 * NP_; i += 256) {
    float v = base[i]; s += v; q += v * v;
  }
  __shared__ float ss[256], qq[256];
  ss[threadIdx.x] = s; qq[threadIdx.x] = q;
  __syncthreads();
  for (int off = 128; off >= 1; off >>= 1) {
    if (threadIdx.x < off) {
      ss[threadIdx.x] += ss[threadIdx.x + off];
      qq[threadIdx.x] += qq[threadIdx.x + off];
    }
    __syncthreads();
  }
  if (threadIdx.x == 0) {
    float cnt = 8.0f * NP_;
    float mean = ss[0] / cnt;
    float var = qq[0] / cnt - mean * mean;
    xstat[bg] = mean;
    xstat[16 + bg] = rsqrtf(var + GEPS_);
  }
}

// ---- K5: vox = W2 * prelu(GN(X1)) + b2 (GN+PReLU fused into B-load) -------
__global__ void k5_vox(const float* __restrict__ w2, const float* __restrict__ b2,
                       const float* __restrict__ x1, const float* __restrict__ g1,
                       const float* __restrict__ beta1, const float* __restrict__ a1,
                       const float* __restrict__ xstat, float* __restrict__ out) {
  int lane = threadIdx.x & 31;
  int mt = threadIdx.x >> 5;
  int b = blockIdx.y;
  int col = blockIdx.x * 16 + (lane & 15);
  int half = lane >> 4;
  int m = mt * 16 + (lane & 15);
  float alpha = a1[0];
  v8f acc = {};
#pragma unroll
  for (int kk = 0; kk < 16; ++kk) {          // K = 64
    int k0 = kk * 4 + 2 * half;
    v2f a, bm;
    a.x = w2[m * C1_ + k0];
    a.y = w2[m * C1_ + k0 + 1];
#pragma unroll
    for (int v = 0; v < 2; ++v) {
      int k = k0 + v;
      int g = k >> 3;
      float raw = x1[((size_t)(b * C1_ + k)) * NP_ + col];
      float xn = (raw - xstat[b * 8 + g]) * xstat[16 + b * 8 + g] * g1[k] + beta1[k];
      xn = xn >= 0.f ? xn : alpha * xn;
      bm[v] = xn;
    }
    acc = wmma4(a, bm, acc);
  }
#pragma unroll
  for (int v = 0; v < 8; ++v) {
    int mm = mt * 16 + v + 8 * half;
    out[((size_t)(b * C1_ + mm)) * NP_ + col] = acc[v] + b2[mm];
  }
}

// ---- K7: y = maxpool_dk( prelu( GN( conv_k(knn) ) ) ) ---------------------
__global__ void k7_ymax(const float* __restrict__ knn, const float* __restrict__ wk,
                        const float* __restrict__ bk, const float* __restrict__ gk,
                        const float* __restrict__ betak, const float* __restrict__ ak,
                        const float* __restrict__ ystat, float* __restrict__ ymax) {
  __shared__ float sk[4][DK_ * 3];
  int c = threadIdx.x;                       // 64 channels
  int pp = threadIdx.y;                      // 4 points per block
  int b = blockIdx.y;
  int p = blockIdx.x * 4 + pp;
  const float* src = knn + ((size_t)(b * NP_) + p) * (DK_ * 3);
  if (c < DK_ * 3) sk[pp][c] = src[c];
  __syncthreads();
  float w0 = wk[c * 3 + 0], w1v = wk[c * 3 + 1], w2v = wk[c * 3 + 2], bc = bk[c];
  int g = c >> 3;
  float mean = ystat[b * 8 + g], rstd = ystat[16 + b * 8 + g];
  float gg = gk[c], be = betak[c], al = ak[0];
  float m = -FINF_;
#pragma unroll
  for (int j = 0; j < DK_; ++j) {
    float yv = w0 * sk[pp][j * 3 + 0] + w1v * sk[pp][j * 3 + 1] + w2v * sk[pp][j * 3 + 2] + bc;
    float yn = (yv - mean) * rstd * gg + be;
    yn = yn >= 0.f ? yn : al * yn;
    m = fmaxf(m, yn);
  }
  ymax[((size_t)(b * C1_ + c)) * NP_ + p] = m;
}

// ---- K8: out += Wo * ymax + bo (WMMA, accumulates onto vox in d_out) ------
__global__ void k8_out(const float* __restrict__ wo, const float* __restrict__ bo,
                       const float* __restrict__ ymax, float* __restrict__ out) {
  int lane = threadIdx.x & 31;
  int mt = threadIdx.x >> 5;
  int b = blockIdx.y;
  int col = blockIdx.x * 16 + (lane & 15);
  int half = lane >> 4;
  int m = mt * 16 + (lane & 15);
  v8f acc = {};
#pragma unroll
  for (int kk = 0; kk < 16; ++kk) {
    int k0 = kk * 4 + 2 * half;
    v2f a, bm;
    a.x = wo[m * C1_ + k0];
    a.y = wo[m * C1_ + k0 + 1];
    const float* yb = ymax + ((size_t)(b * C1_ + k0)) * NP_ + col;
    bm.x = yb[0]; bm.y = yb[NP_];
    acc = wmma4(a, bm, acc);
  }
#pragma unroll
  for (int v = 0; v < 8; ++v) {
    int mm = mt * 16 + v + 8 * half;
    size_t idx = ((size_t)(b * C1_ + mm)) * NP_ + col;
    out[idx] = out[idx] + acc[v] + bo[mm];
  }
}

extern "C" void kernel_launch(void* const* d_in, const int* in_sizes, int n_in,
                              void* d_out, int out_size, void* d_ws, size_t ws_size,
                              hipStream_t stream) {
  (void)in_sizes; (void)n_in; (void)out_size; (void)ws_size;
  const float* coords = (const float*)d_in[0];
  const float* adf    = (const float*)d_in[1];
  const float* corr   = (const float*)d_in[2];
  const float* xy2    = (const float*)d_in[3];
  const float* w1  = (const float*)d_in[4];
  const float* b1  = (const float*)d_in[5];
  const float* g1  = (const float*)d_in[6];
  const float* bt1 = (const float*)d_in[7];
  const float* a1  = (const float*)d_in[8];
  const float* w2  = (const float*)d_in[9];
  const float* b2  = (const float*)d_in[10];
  const float* wk  = (const float*)d_in[11];
  const float* bk  = (const float*)d_in[12];
  const float* gk  = (const float*)d_in[13];
  const float* btk = (const float*)d_in[14];
  const float* ak  = (const float*)d_in[15];
  const float* wo  = (const float*)d_in[16];
  const float* bo  = (const float*)d_in[17];
  const int* scale = (const int*)d_in[19];

  float* ws  = (float*)d_ws;
  float* out = (float*)d_out;
  float* atom  = ws + OFF_ATOM;
  float* xstat = ws + OFF_XSTAT;
  float* ystat = ws + OFF_YSTAT;
  float* ypart = ws + OFF_YPART;
  float* feat  = ws + OFF_FEAT;
  float* knn   = ws + OFF_KNN;
  float* x1    = ws + OFF_X1;
  float* ymax  = ws + OFF_YMAX;

  hipMemsetAsync(atom, 0, 64 * sizeof(float), stream);
  k1_rmax <<<dim3(NP_ / 256, B_), 256, 0, stream>>>(adf, atom);
  k2a_feat<<<dim3(NP_ / 256, B_), 256, 0, stream>>>(coords, adf, corr, xy2, scale, atom, feat);
  k2b_knn <<<dim3(NP_ / WPB_, B_), dim3(32, WPB_), 0, stream>>>(coords, corr, xy2, wk, bk, knn, ypart);
  k6_ystats<<<1, 32, 0, stream>>>(ypart, ystat);
  k3_gemm1<<<dim3(NP_ / 16, B_), 128, 0, stream>>>(w1, b1, feat, x1);
  k4_xstats<<<16, 256, 0, stream>>>(x1, xstat);
  k5_vox  <<<dim3(NP_ / 16, B_), 128, 0, stream>>>(w2, b2, x1, g1, bt1, a1, xstat, out);
  k7_ymax <<<dim3(NP_ / 4, B_), dim3(64, 4), 0, stream>>>(knn, wk, bk, gk, btk, ak, ystat, ymax);
  k8_out  <<<dim3(NP_ / 16, B_), 128, 0, stream>>>(wo, bo, ymax, out);
}